// HieNNClassifier_78288663872087
// MI455X (gfx1250) — compile-verified
//
#include <hip/hip_runtime.h>
#include <hip/hip_bf16.h>
#include <math.h>

// Shapes from the reference
#define BB 64
#define SS 1024
#define EE 256
#define HH 256
#define CC 20

#define KCHUNK 64          // K rows of W1 staged per LDS chunk (64 KB)
#define SESTRIDE 260       // padded row stride for sentence-embedding tile

typedef float v2f __attribute__((ext_vector_type(2)));
typedef float v8f __attribute__((ext_vector_type(8)));

#if __has_builtin(__builtin_amdgcn_global_load_async_to_lds_b32)
#define HAVE_ASYNC_LDS 1
#else
#define HAVE_ASYNC_LDS 0
#endif

// Builtin signature (from hipcc diagnostic): (as1 int*, as3 int*, Imm i32, Imm i32)
typedef __attribute__((address_space(1))) int* gas1_i32p;
typedef __attribute__((address_space(3))) int* las3_i32p;

__device__ __forceinline__ void wait_asynccnt0() {
#if __has_builtin(__builtin_amdgcn_s_wait_asynccnt)
    __builtin_amdgcn_s_wait_asynccnt(0);
#else
    asm volatile("s_wait_asynccnt 0x0" ::: "memory");
#endif
}

// ---------------------------------------------------------------------------
// Kernel 1: h = tanh(emb[batch_x] @ W1 + b1)
// fp32 WMMA 16x16x4. Wave: 16 token rows x 256 cols (16 acc tiles).
// W1 staged in LDS via swizzled ASYNC copy (repack happens in the per-lane LDS
// destination address), so each B fragment is one ds_load_b64; fragments for a
// whole K-step are batched so the loop pays one s_wait per 16 WMMAs.
// Block = 8 waves = 128 rows. Grid = 65536/128 = 512.
// ---------------------------------------------------------------------------
__global__ __launch_bounds__(256) void k_embed_mlp1(
    const int* __restrict__ batch_x,
    const float* __restrict__ emb,
    const float* __restrict__ W1,   // [E][H] row-major
    const float* __restrict__ b1,   // [H]
    float* __restrict__ h_out)      // [B*S][H]
{
    // repacked chunk: buf[(kr>>1)*512 + n*2 + (kr&1)] = W1[kc+kr][n]
    __shared__ float w1buf[(KCHUNK / 2) * 512];   // 16384 floats = 64 KB

    const int wave = threadIdx.x >> 5;
    const int lane = threadIdx.x & 31;
    const int row0 = (blockIdx.x * 8 + wave) * 16;

    const int m     = lane & 15;    // A row / B-C column-within-tile
    const int khalf = lane >> 4;    // 0: K={0,1}  1: K={2,3}

    const int tok = batch_x[row0 + m];
    const float* __restrict__ arow = emb + (long long)tok * EE;

    v8f acc[16];
    v8f zero = {};
#pragma unroll
    for (int t = 0; t < 16; ++t) acc[t] = zero;

    for (int kc = 0; kc < EE; kc += KCHUNK) {
        __syncthreads();            // protect previous chunk's readers
        {
            const int n = threadIdx.x;                 // column owned by thread
            const float* __restrict__ src = W1 + (long long)kc * HH + n;
#if HAVE_ASYNC_LDS
            // swizzled async copy: repack encoded in per-lane LDS dest address
#pragma unroll 8
            for (int kr = 0; kr < KCHUNK; ++kr) {
                __builtin_amdgcn_global_load_async_to_lds_b32(
                    (gas1_i32p)(src + kr * HH),
                    (las3_i32p)&w1buf[(kr >> 1) * 512 + n * 2 + (kr & 1)],
                    0, 0);
            }
            wait_asynccnt0();       // async writes landed in LDS before barrier
#else
#pragma unroll 4
            for (int kr = 0; kr < KCHUNK; ++kr) {
                w1buf[(kr >> 1) * 512 + n * 2 + (kr & 1)] = src[kr * HH];
            }
#endif
        }
        __syncthreads();

        // prefetch next chunk toward L2 while we compute (global_prefetch_b8)
        if (kc + KCHUNK < EE) {
            const float* np = W1 + (long long)(kc + KCHUNK) * HH + threadIdx.x * 64;
            __builtin_prefetch(np, 0, 1);
        }

#pragma unroll 2
        for (int k = 0; k < KCHUNK; k += 4) {
            const int kk = k + khalf * 2;       // this lane's first K row
            v2f a = *(const v2f*)&arow[kc + kk];
            const float* __restrict__ bp = &w1buf[(kk >> 1) * 512];

            // batch all 16 B fragments first -> clause of ds_load_2addr_b64,
            // single s_wait_dscnt before the WMMA burst
            v2f bf[16];
#pragma unroll
            for (int t = 0; t < 16; ++t)
                bf[t] = *(const v2f*)(bp + (t * 16 + m) * 2);
#pragma unroll
            for (int t = 0; t < 16; ++t)
                acc[t] = __builtin_amdgcn_wmma_f32_16x16x4_f32(
                    false, a, false, bf[t], (short)0, acc[t], false, false);
        }
    }

    // Epilogue: C layout: vgpr j -> row (j + 8*khalf), col = t*16 + m
#pragma unroll
    for (int t = 0; t < 16; ++t) {
        const int c = t * 16 + m;
        const float bias = b1[c];
#pragma unroll
        for (int j = 0; j < 8; ++j) {
            const int r = row0 + j + 8 * khalf;
            h_out[(long long)r * HH + c] = tanhf(acc[t][j] + bias);
        }
    }
}

// ---------------------------------------------------------------------------
// Kernel 2: boundary scan -> contiguous segment means (16 at a time into LDS)
// -> WMMA GEMM with W2 -> tanh + masked row-accumulate into doc (ds_add_f32)
// -> W3 -> log_softmax. One block (256 thr / 8 waves) per batch.
// ---------------------------------------------------------------------------
__global__ __launch_bounds__(256) void k_segment_docs(
    const int* __restrict__ batch_x,
    const float* __restrict__ h,     // [B*S][H]
    const float* __restrict__ W2,    // [H][H]
    const float* __restrict__ b2,    // [H]
    const float* __restrict__ W3,    // [H][C]
    const float* __restrict__ b3,    // [C]
    float* __restrict__ out)         // [B][C]
{
    __shared__ int   bpos[SS];             // boundary positions
    __shared__ int   scan[256];
    __shared__ float se[16 * SESTRIDE];    // 16 sentence embeddings (padded rows)
    __shared__ float doc[HH];              // doc accumulator
    __shared__ float cat[32];
    __shared__ float red[2];

    const int b = blockIdx.x;
    const int t = threadIdx.x;
    const int* __restrict__ xb = batch_x + b * SS;

    // --- boundary flags (4 tokens/thread) + block inclusive scan ---
    int f[4];
    int cnt = 0;
#pragma unroll
    for (int i = 0; i < 4; ++i) { f[i] = (xb[t * 4 + i] == 1); cnt += f[i]; }
    scan[t] = cnt;
    __syncthreads();
    for (int off = 1; off < 256; off <<= 1) {
        int v   = scan[t];
        int add = (t >= off) ? scan[t - off] : 0;
        __syncthreads();
        scan[t] = v + add;
        __syncthreads();
    }
    const int nb = scan[255];              // doc_lens[b] (>=1)
    int base = scan[t] - cnt;
#pragma unroll
    for (int i = 0; i < 4; ++i) {
        if (f[i]) bpos[base++] = t * 4 + i;
    }
    doc[t] = 0.0f;
    __syncthreads();

    const int lane  = t & 31;
    const int wave  = t >> 5;
    const int m     = lane & 15;
    const int khalf = lane >> 4;
    const int t0    = wave * 2;            // this wave's two column tiles
    const int t1    = wave * 2 + 1;
    const int c0    = t0 * 16 + m;
    const int c1    = t1 * 16 + m;

    for (int g = 0; g < nb; g += 16) {
        // --- phase A: means of up to 16 contiguous segments -> se[][] ---
        for (int i = 0; i < 16; ++i) {
            const int s = g + i;
            float v = 0.0f;
            if (s < nb) {
                const int start = (s == 0) ? 0 : bpos[s - 1] + 1;
                const int end   = bpos[s];
                const float* __restrict__ hp =
                    h + ((long long)b * SS + start) * HH + t;
                float a = 0.0f;
                for (int j = start; j <= end; ++j, hp += HH) a += *hp;
                v = a / (float)(end - start + 1);
            }
            se[i * SESTRIDE + t] = v;      // invalid rows zero-padded
        }
        __syncthreads();

        // --- phase B: 16x256 = tanh(se @ W2 + b2) on the matrix pipe ---
        v8f acc0 = {}, acc1 = {};
        for (int k = 0; k < HH; k += 4) {
            const int kk = k + khalf * 2;
            v2f a = *(const v2f*)&se[m * SESTRIDE + kk];   // ds_load_b64
            const float* __restrict__ w0 = W2 + (long long)kk * HH;
            v2f bf0, bf1;
            bf0.x = w0[c0];  bf0.y = w0[HH + c0];
            bf1.x = w0[c1];  bf1.y = w0[HH + c1];
            acc0 = __builtin_amdgcn_wmma_f32_16x16x4_f32(
                false, a, false, bf0, (short)0, acc0, false, false);
            acc1 = __builtin_amdgcn_wmma_f32_16x16x4_f32(
                false, a, false, bf1, (short)0, acc1, false, false);
        }

        // --- phase C: tanh + bias, mask invalid rows, accumulate into doc ---
        {
            const float bias0 = b2[c0];
            const float bias1 = b2[c1];
            float p0 = 0.0f, p1 = 0.0f;
#pragma unroll
            for (int j = 0; j < 8; ++j) {
                const int i = j + 8 * khalf;   // segment row within group
                if (g + i < nb) {
                    p0 += tanhf(acc0[j] + bias0);
                    p1 += tanhf(acc1[j] + bias1);
                }
            }
            atomicAdd(&doc[c0], p0);           // ds_add_f32
            atomicAdd(&doc[c1], p1);
        }
        __syncthreads();                       // before se[] is overwritten
    }

    const float dv = doc[t] / (float)nb;
    __syncthreads();
    doc[t] = dv;
    __syncthreads();

    // --- cat_space = doc @ W3 + b3, log_softmax over C=20 ---
    if (t < CC) {
        float cv = b3[t];
        for (int k = 0; k < HH; ++k) cv = fmaf(doc[k], W3[k * CC + t], cv);
        cat[t] = cv;
    }
    __syncthreads();
    if (t == 0) {
        float mx = cat[0];
        for (int i = 1; i < CC; ++i) mx = fmaxf(mx, cat[i]);
        float sum = 0.0f;
        for (int i = 0; i < CC; ++i) sum += __expf(cat[i] - mx);
        red[0] = mx;
        red[1] = logf(sum);
    }
    __syncthreads();
    if (t < CC) out[b * CC + t] = cat[t] - red[0] - red[1];
}

// ---------------------------------------------------------------------------
// Inputs (setup_inputs order):
// 0 batch_x(int B*S) 1 batch_lens(int B) 2 emb(f32 V*E) 3 W1 4 b1 5 W2 6 b2 7 W3 8 b3
// Output: B*C f32. Workspace: h = B*S*H f32 (64 MiB).
// ---------------------------------------------------------------------------
extern "C" void kernel_launch(void* const* d_in, const int* in_sizes, int n_in,
                              void* d_out, int out_size, void* d_ws, size_t ws_size,
                              hipStream_t stream)
{
    const int*   batch_x = (const int*)  d_in[0];
    const float* emb     = (const float*)d_in[2];
    const float* W1      = (const float*)d_in[3];
    const float* b1      = (const float*)d_in[4];
    const float* W2      = (const float*)d_in[5];
    const float* b2      = (const float*)d_in[6];
    const float* W3      = (const float*)d_in[7];
    const float* b3      = (const float*)d_in[8];

    float* h = (float*)d_ws;                    // B*S*H floats = 64 MiB
    const int n_rows = BB * SS;                 // 65536

    k_embed_mlp1<<<n_rows / 128, 256, 0, stream>>>(batch_x, emb, W1, b1, h);
    k_segment_docs<<<BB, 256, 0, stream>>>(batch_x, h, W2, b2, W3, b3, (float*)d_out);
}